// RelGCN_38628935860967
// MI455X (gfx1250) — compile-verified
//
#include <hip/hip_runtime.h>
#include <hip/hip_bf16.h>

typedef __attribute__((ext_vector_type(2))) float v2f;
typedef __attribute__((ext_vector_type(4))) float v4f;
typedef __attribute__((ext_vector_type(8))) float v8f;

#define F_IN 128
#define HDIM 128
#define CDIM 16
#define RREL 5
#define KR   640          // R * 128 (relation part of concatenated K)
#define KTOT 768          // KR + 128 (root part)

// ---------------- edge-count kernel: cnt[dst*R+et] += 1 ----------------
__global__ void count_edges(const int* __restrict__ dst,
                            const int* __restrict__ et,
                            float* __restrict__ cnt, int E) {
  int e = blockIdx.x * blockDim.x + threadIdx.x;
  if (e < E) atomicAdd(&cnt[(size_t)dst[e] * RREL + et[e]], 1.0f);
}

// ------ scatter kernel: agg[(dst,et),f] += ew * feat[src,f], 4 feats/thread ------
__global__ void scatter_edges(const float* __restrict__ feat,
                              const int* __restrict__ src,
                              const int* __restrict__ dst,
                              const float* __restrict__ ew,
                              const int* __restrict__ et,
                              float* __restrict__ agg, int E) {
  unsigned i = blockIdx.x * blockDim.x + threadIdx.x;
  unsigned total = (unsigned)E << 5;          // E * 32 quads
  if (i >= total) return;
  int e = i >> 5, f4 = (i & 31) << 2;
  float w = ew[e];
  v4f m = *(const v4f*)(feat + (size_t)src[e] * 128 + f4);
  float* p = agg + ((size_t)dst[e] * RREL + et[e]) * 128 + f4;
  atomicAdd(p + 0, w * m.x);
  atomicAdd(p + 1, w * m.y);
  atomicAdd(p + 2, w * m.z);
  atomicAdd(p + 3, w * m.w);
}

// -------- pack [W (KRxN) ; root (128xN)] into f32 WMMA B-fragment layout --------
// v2f index (ks*ntiles + j)*32 + lane ; lane<16 holds rows k,k+1 ; lane>=16 rows k+2,k+3
__global__ void pack_weights(const float* __restrict__ W,
                             const float* __restrict__ root,
                             float* __restrict__ pack, int Ncols) {
  int idx = blockIdx.x * blockDim.x + threadIdx.x;
  int total = KTOT * Ncols;
  if (idx >= total) return;
  int k = idx / Ncols, n = idx % Ncols;
  float v = (k < KR) ? W[(size_t)k * Ncols + n]
                     : root[(size_t)(k - KR) * Ncols + n];
  int ks = k >> 2, kk = k & 3;
  int j = n >> 4;
  int lane = ((kk >> 1) << 4) | (n & 15);
  int pair = kk & 1;
  int ntiles = Ncols >> 4;
  pack[((((size_t)ks * ntiles + j) * 32 + lane) << 1) + pair] = v;
}

// --------- layer-1 GEMM: h = relu([agg/cnt | x] @ [W1;root1] + b1) ---------
// one wave = 16 rows x 128 cols; per-relation sub-loops keep the scale invariant;
// B fragments batched into an array so the 8 loads issue as one clause.
__global__ void gemm_layer1(const float* __restrict__ agg,
                            const float* __restrict__ cnt,
                            const float* __restrict__ x,
                            const float* __restrict__ pack,
                            const float* __restrict__ bias,
                            float* __restrict__ hout, int N) {
  int wave = (blockIdx.x * blockDim.x + threadIdx.x) >> 5;
  int lane = threadIdx.x & 31;
  int row0 = wave << 4;
  if (row0 >= N) return;                    // wave-uniform: EXEC stays all-1s for WMMA
  int r  = row0 + (lane & 15);
  int hi = lane >> 4;
  const float* aggRow = agg + (size_t)r * KR;
  const float* xRow   = x   + (size_t)r * 128;
  const v2f*   pk     = (const v2f*)pack;
  v8f acc[8] = {};
  // relation part: 5 sub-loops of 32 k-steps, loop-invariant 1/cnt scale
  for (int rel = 0; rel < RREL; ++rel) {
    float s = 1.0f / fmaxf(cnt[(size_t)r * RREL + rel], 1.0f);
    const float* aRel = aggRow + rel * 128 + (hi << 1);
    const v2f*   bRel = pk + (size_t)(rel * 32) * 8 * 32 + lane;
    for (int t = 0; t < 32; ++t) {
      const v2f* pb = bRel + (size_t)t * 8 * 32;
      v2f a = *(const v2f*)(aRel + (t << 2));
      v2f b[8];
#pragma unroll
      for (int j = 0; j < 8; ++j) b[j] = pb[(size_t)j * 32];
      a.x *= s; a.y *= s;
#pragma unroll
      for (int j = 0; j < 8; ++j)
        acc[j] = __builtin_amdgcn_wmma_f32_16x16x4_f32(
            false, a, false, b[j], (short)0, acc[j], false, false);
    }
  }
  // root part: 32 k-steps over x, no scaling
  {
    const float* aRt = xRow + (hi << 1);
    const v2f*   bRt = pk + (size_t)160 * 8 * 32 + lane;
    for (int t = 0; t < 32; ++t) {
      const v2f* pb = bRt + (size_t)t * 8 * 32;
      v2f a = *(const v2f*)(aRt + (t << 2));
      v2f b[8];
#pragma unroll
      for (int j = 0; j < 8; ++j) b[j] = pb[(size_t)j * 32];
#pragma unroll
      for (int j = 0; j < 8; ++j)
        acc[j] = __builtin_amdgcn_wmma_f32_16x16x4_f32(
            false, a, false, b[j], (short)0, acc[j], false, false);
    }
  }
  int nlo = lane & 15, moff = hi << 3;
#pragma unroll
  for (int j = 0; j < 8; ++j) {
    int col = (j << 4) + nlo;
    float bv = bias[col];
#pragma unroll
    for (int v = 0; v < 8; ++v) {           // C layout: VGPR v -> M = v + 8*hi
      int rr = row0 + moff + v;
      hout[(size_t)rr * HDIM + col] = fmaxf(acc[j][v] + bv, 0.0f);
    }
  }
}

// --- layer-2 GEMM + fused log_softmax: out = logsoftmax([agg/cnt | h] @ [W2;root2] + b2) ---
// 8 k-steps of A and B fragments batched per iteration -> 16 loads in flight, then 8 WMMAs.
__global__ void gemm_layer2(const float* __restrict__ agg,
                            const float* __restrict__ cnt,
                            const float* __restrict__ hfeat,
                            const float* __restrict__ pack,
                            const float* __restrict__ bias,
                            float* __restrict__ out, int N) {
  int wave = (blockIdx.x * blockDim.x + threadIdx.x) >> 5;
  int lane = threadIdx.x & 31;
  int row0 = wave << 4;
  if (row0 >= N) return;
  int r  = row0 + (lane & 15);
  int hi = lane >> 4;
  const float* aggRow = agg   + (size_t)r * KR;
  const float* hRow   = hfeat + (size_t)r * 128;
  const v2f*   pk     = (const v2f*)pack;
  v8f acc = {};
  for (int rel = 0; rel < RREL; ++rel) {
    float s = 1.0f / fmaxf(cnt[(size_t)r * RREL + rel], 1.0f);
    const float* aRel = aggRow + rel * 128 + (hi << 1);
    const v2f*   bRel = pk + (size_t)(rel * 32) * 32 + lane;
    for (int t = 0; t < 32; t += 8) {
      v2f a[8], b[8];
#pragma unroll
      for (int u = 0; u < 8; ++u) a[u] = *(const v2f*)(aRel + ((t + u) << 2));
#pragma unroll
      for (int u = 0; u < 8; ++u) b[u] = bRel[(size_t)(t + u) * 32];
#pragma unroll
      for (int u = 0; u < 8; ++u) { a[u].x *= s; a[u].y *= s; }
#pragma unroll
      for (int u = 0; u < 8; ++u)
        acc = __builtin_amdgcn_wmma_f32_16x16x4_f32(
            false, a[u], false, b[u], (short)0, acc, false, false);
    }
  }
  {
    const float* aRt = hRow + (hi << 1);
    const v2f*   bRt = pk + (size_t)160 * 32 + lane;
    for (int t = 0; t < 32; t += 8) {
      v2f a[8], b[8];
#pragma unroll
      for (int u = 0; u < 8; ++u) a[u] = *(const v2f*)(aRt + ((t + u) << 2));
#pragma unroll
      for (int u = 0; u < 8; ++u) b[u] = bRt[(size_t)(t + u) * 32];
#pragma unroll
      for (int u = 0; u < 8; ++u)
        acc = __builtin_amdgcn_wmma_f32_16x16x4_f32(
            false, a[u], false, b[u], (short)0, acc, false, false);
    }
  }
  // each output row lives in one 16-lane half -> butterfly log_softmax over 16 classes
  int col = lane & 15;
  float bv = bias[col];
#pragma unroll
  for (int v = 0; v < 8; ++v) {
    float val = acc[v] + bv;
    float m = val;
#pragma unroll
    for (int d = 1; d < 16; d <<= 1) m = fmaxf(m, __shfl_xor(m, d, 32));
    float p = expf(val - m);
    float s = p;
#pragma unroll
    for (int d = 1; d < 16; d <<= 1) s += __shfl_xor(s, d, 32);
    int rr = row0 + (hi << 3) + v;
    out[(size_t)rr * CDIM + col] = val - m - logf(s);
  }
}

extern "C" void kernel_launch(void* const* d_in, const int* in_sizes, int n_in,
                              void* d_out, int out_size, void* d_ws, size_t ws_size,
                              hipStream_t stream) {
  const float* x     = (const float*)d_in[0];
  const int*   ei    = (const int*)d_in[1];   // [2, E]
  const float* ew    = (const float*)d_in[2];
  const int*   et    = (const int*)d_in[3];
  const float* W1    = (const float*)d_in[4];
  const float* root1 = (const float*)d_in[5];
  const float* b1    = (const float*)d_in[6];
  const float* W2    = (const float*)d_in[7];
  const float* root2 = (const float*)d_in[8];
  const float* b2    = (const float*)d_in[9];

  int N = in_sizes[0] / F_IN;
  int E = in_sizes[2];
  const int* src = ei;
  const int* dst = ei + E;

  // workspace layout (256B aligned): cnt | agg (shared by both layers) | h | pack1 | pack2
  char* ws = (char*)d_ws;
  size_t off = 0;
  auto take = [&](size_t bytes) {
    char* p = ws + off;
    off = (off + bytes + 255) & ~(size_t)255;
    return p;
  };
  float* cnt   = (float*)take((size_t)N * RREL * sizeof(float));
  float* agg   = (float*)take((size_t)N * RREL * 128 * sizeof(float));
  float* hbuf  = (float*)take((size_t)N * HDIM * sizeof(float));
  float* pack1 = (float*)take((size_t)KTOT * HDIM * sizeof(float));
  float* pack2 = (float*)take((size_t)KTOT * CDIM * sizeof(float));

  hipMemsetAsync(cnt, 0, (size_t)N * RREL * sizeof(float), stream);
  hipMemsetAsync(agg, 0, (size_t)N * RREL * 128 * sizeof(float), stream);

  count_edges<<<(E + 255) / 256, 256, 0, stream>>>(dst, et, cnt, E);
  pack_weights<<<(KTOT * HDIM + 255) / 256, 256, 0, stream>>>(W1, root1, pack1, HDIM);
  pack_weights<<<(KTOT * CDIM + 255) / 256, 256, 0, stream>>>(W2, root2, pack2, CDIM);

  unsigned quads = (unsigned)E * 32u;
  scatter_edges<<<(quads + 255u) / 256u, 256, 0, stream>>>(x, src, dst, ew, et, agg, E);

  int waves   = (N + 15) / 16;
  int gblocks = (waves * 32 + 255) / 256;
  gemm_layer1<<<gblocks, 256, 0, stream>>>(agg, cnt, x, pack1, b1, hbuf, N);

  hipMemsetAsync(agg, 0, (size_t)N * RREL * 128 * sizeof(float), stream);
  scatter_edges<<<(quads + 255u) / 256u, 256, 0, stream>>>(hbuf, src, dst, ew, et, agg, E);

  gemm_layer2<<<gblocks, 256, 0, stream>>>(agg, cnt, hbuf, pack2, b2, (float*)d_out, N);
}